// RadarDopSparseProcessor_22119081575168
// MI455X (gfx1250) — compile-verified
//
#include <hip/hip_runtime.h>
#include <hip/hip_bf16.h>
#include <stdint.h>

// ---------------- problem geometry (compile-time constants) ----------------
#define BATCH 4
#define ZD 40
#define YD 320
#define XD 320
#define NPER (ZD * YD * XD)          // 4,096,000 voxels / sample
#define KPAD (NPER / 10 + 64)        // 409,664 padded rows / sample
#define RANK0 3686399u               // floor(0.9*(N-1))
#define QFRAC 0.1f                   // 0.9*(N-1) - RANK0
#define POW_SCALE 1.0e13f
#define DOP_OFFSET 1.9326f

#define CHUNK 4096                   // elements handled per workgroup per pass
#define BLOCKS_PER_SAMPLE (NPER / CHUNK)   // 1000 (exact)
#define BLOCKS_PAD 1024

typedef float v2f __attribute__((ext_vector_type(2)));
typedef float v8f __attribute__((ext_vector_type(8)));

// ============================================================================
// Radix-select level 1: histogram of float-bit top 12 bits (bits[31:20]).
// Inputs are uniform [0,1) -> non-negative -> bit pattern is order-preserving.
// LDS-privatized histogram (ds_add), merged into global with atomics.
// ============================================================================
__global__ void k_hist_l1(const float* __restrict__ cube,
                          unsigned int* __restrict__ gh1) {
  __shared__ unsigned int h[4096];
  const int b = blockIdx.y;
  const int tid = threadIdx.x;
  for (int i = tid; i < 4096; i += 256) h[i] = 0u;
  __syncthreads();

  const float4* src = (const float4*)(cube + (size_t)b * NPER +
                                      (size_t)blockIdx.x * CHUNK);
  for (int r = 0; r < 4; ++r) {
    float4 v = src[r * 256 + tid];
    atomicAdd(&h[__float_as_uint(v.x) >> 20], 1u);
    atomicAdd(&h[__float_as_uint(v.y) >> 20], 1u);
    atomicAdd(&h[__float_as_uint(v.z) >> 20], 1u);
    atomicAdd(&h[__float_as_uint(v.w) >> 20], 1u);
  }
  __syncthreads();
  unsigned int* g = gh1 + (size_t)b * 4096;
  for (int i = tid; i < 4096; i += 256)
    if (h[i]) atomicAdd(&g[i], h[i]);
}

// One thread per (sample, rank j) walks 4096 bins to locate the bin holding
// rank (RANK0 + j) and the residual rank inside that bin.
__global__ void k_pick_l1(const unsigned int* __restrict__ gh1,
                          unsigned int* __restrict__ selP1,
                          unsigned int* __restrict__ selR1) {
  const int t = threadIdx.x;
  if (t >= BATCH * 2) return;
  const int b = t >> 1, j = t & 1;
  const unsigned int r = RANK0 + (unsigned int)j;
  const unsigned int* h = gh1 + (size_t)b * 4096;
  unsigned int cum = 0, pfx = 4095, rem = 0;
  for (int i = 0; i < 4096; ++i) {
    unsigned int c = h[i];
    if (cum + c > r) { pfx = (unsigned int)i; rem = r - cum; break; }
    cum += c;
  }
  selP1[t] = pfx;
  selR1[t] = rem;
}

// ============================================================================
// Level 2: histogram of bits[19:8] restricted to the two candidate 12-bit
// prefixes (one per rank). Two 4096-bin LDS histograms (32 KB of 320 KB WGP).
// ============================================================================
__global__ void k_hist_l2(const float* __restrict__ cube,
                          const unsigned int* __restrict__ selP1,
                          unsigned int* __restrict__ gh2) {
  __shared__ unsigned int h[2][4096];
  const int b = blockIdx.y;
  const int tid = threadIdx.x;
  for (int i = tid; i < 8192; i += 256) ((unsigned int*)h)[i] = 0u;
  const unsigned int p0 = selP1[b * 2 + 0];
  const unsigned int p1 = selP1[b * 2 + 1];
  __syncthreads();

  const float4* src = (const float4*)(cube + (size_t)b * NPER +
                                      (size_t)blockIdx.x * CHUNK);
  for (int r = 0; r < 4; ++r) {
    float4 v = src[r * 256 + tid];
#define P2(f)                                                     \
    { unsigned int kk = __float_as_uint(f);                       \
      unsigned int top = kk >> 20, bin = (kk >> 8) & 0xFFFu;      \
      if (top == p0) atomicAdd(&h[0][bin], 1u);                   \
      if (top == p1) atomicAdd(&h[1][bin], 1u); }
    P2(v.x) P2(v.y) P2(v.z) P2(v.w)
#undef P2
  }
  __syncthreads();
  unsigned int* g = gh2 + (size_t)b * 8192;
  for (int i = tid; i < 8192; i += 256) {
    unsigned int c = ((unsigned int*)h)[i];
    if (c) atomicAdd(&g[i], c);
  }
}

__global__ void k_pick_l2(const unsigned int* __restrict__ gh2,
                          const unsigned int* __restrict__ selR1,
                          unsigned int* __restrict__ selP2,
                          unsigned int* __restrict__ selR2) {
  const int t = threadIdx.x;
  if (t >= BATCH * 2) return;
  const unsigned int r = selR1[t];
  const unsigned int* h = gh2 + (size_t)t * 4096;
  unsigned int cum = 0, pfx = 4095, rem = 0;
  for (int i = 0; i < 4096; ++i) {
    unsigned int c = h[i];
    if (cum + c > r) { pfx = (unsigned int)i; rem = r - cum; break; }
    cum += c;
  }
  selP2[t] = pfx;
  selR2[t] = rem;
}

// ============================================================================
// Level 3: histogram of low byte restricted to the 24-bit candidate prefixes.
// ============================================================================
__global__ void k_hist_l3(const float* __restrict__ cube,
                          const unsigned int* __restrict__ selP1,
                          const unsigned int* __restrict__ selP2,
                          unsigned int* __restrict__ gh3) {
  __shared__ unsigned int h[2][256];
  const int b = blockIdx.y;
  const int tid = threadIdx.x;
  for (int i = tid; i < 512; i += 256) ((unsigned int*)h)[i] = 0u;
  const unsigned int p0 = (selP1[b * 2 + 0] << 12) | selP2[b * 2 + 0];
  const unsigned int p1 = (selP1[b * 2 + 1] << 12) | selP2[b * 2 + 1];
  __syncthreads();

  const float4* src = (const float4*)(cube + (size_t)b * NPER +
                                      (size_t)blockIdx.x * CHUNK);
  for (int r = 0; r < 4; ++r) {
    float4 v = src[r * 256 + tid];
#define P3(f)                                                     \
    { unsigned int kk = __float_as_uint(f);                       \
      unsigned int top = kk >> 8;                                 \
      if (top == p0) atomicAdd(&h[0][kk & 0xFFu], 1u);            \
      if (top == p1) atomicAdd(&h[1][kk & 0xFFu], 1u); }
    P3(v.x) P3(v.y) P3(v.z) P3(v.w)
#undef P3
  }
  __syncthreads();
  unsigned int* g = gh3 + (size_t)b * 512;
  for (int i = tid; i < 512; i += 256) {
    unsigned int c = ((unsigned int*)h)[i];
    if (c) atomicAdd(&g[i], c);
  }
}

// Resolve the final byte -> exact order statistics -> interpolated threshold.
__global__ void k_pick_l3_thr(const unsigned int* __restrict__ gh3,
                              const unsigned int* __restrict__ selP1,
                              const unsigned int* __restrict__ selP2,
                              const unsigned int* __restrict__ selR2,
                              float* __restrict__ thr) {
  __shared__ float sv[BATCH * 2];
  const int t = threadIdx.x;
  if (t < BATCH * 2) {
    const unsigned int* h = gh3 + (size_t)t * 256;
    const unsigned int r = selR2[t];
    unsigned int cum = 0, byte_ = 255;
    for (int i = 0; i < 256; ++i) {
      unsigned int c = h[i];
      if (cum + c > r) { byte_ = (unsigned int)i; break; }
      cum += c;
    }
    unsigned int bits = (selP1[t] << 20) | (selP2[t] << 8) | byte_;
    sv[t] = __uint_as_float(bits);
  }
  __syncthreads();
  if (t < BATCH) thr[t] = sv[t * 2] + QFRAC * (sv[t * 2 + 1] - sv[t * 2]);
}

// ============================================================================
// Per-block survivor counts (order-insensitive, float4 streaming).
// ============================================================================
__global__ void k_count(const float* __restrict__ cube,
                        const float* __restrict__ thr,
                        unsigned int* __restrict__ blockCnt) {
  __shared__ unsigned int red[256];
  const int b = blockIdx.y;
  const int tid = threadIdx.x;
  const float t = thr[b];
  const float4* src = (const float4*)(cube + (size_t)b * NPER +
                                      (size_t)blockIdx.x * CHUNK);
  unsigned int c = 0;
  for (int r = 0; r < 4; ++r) {
    float4 v = src[r * 256 + tid];
    c += (v.x > t) + (v.y > t) + (v.z > t) + (v.w > t);
  }
  red[tid] = c;
  __syncthreads();
  for (int s = 128; s > 0; s >>= 1) {
    if (tid < s) red[tid] += red[tid + s];
    __syncthreads();
  }
  if (tid == 0) blockCnt[(size_t)b * BLOCKS_PAD + blockIdx.x] = red[0];
}

// ============================================================================
// WMMA matrix-scan: exclusive prefix sum of 1024 block counts per sample.
// Inclusive-scan-within-row via M(16x16) x U(upper-tri ones), done as 4
// chained V_WMMA_F32_16X16X4_F32 (K accumulation); cross-row carries are a
// trivial 64-step serial fix-up. Counts <= 4096, totals <= 4.1M: exact in f32.
// One wave per sample (EXEC all-ones as WMMA requires).
// Operand layouts per ISA 7.12.2: A 16x4 f32 -> lanes 0-15 {K0,K1}, lanes
// 16-31 {K2,K3}; B 4x16 symmetric; C/D = documented 8-VGPR striping.
// ============================================================================
__global__ void k_scan_wmma(const unsigned int* __restrict__ blockCnt,
                            unsigned int* __restrict__ blockOff,
                            unsigned int* __restrict__ totalCnt) {
  __shared__ float vals[BLOCKS_PAD];
  __shared__ float incl[BLOCKS_PAD];
  __shared__ float carry[64];
  const int b = blockIdx.x;
  const int lane = threadIdx.x;            // 0..31, one full wave32
  const unsigned int* src = blockCnt + (size_t)b * BLOCKS_PAD;
  for (int i = lane; i < BLOCKS_PAD; i += 32) vals[i] = (float)src[i];
  __syncthreads();

  const int half = lane >> 4;              // 0: lanes 0-15, 1: lanes 16-31
  const int l15 = lane & 15;

  for (int t = 0; t < 4; ++t) {
    const float* M = &vals[t * 256];       // M[m][n] = vals[t*256 + m*16 + n]
    v8f acc = {};
#pragma unroll
    for (int g = 0; g < 4; ++g) {
      const int k0 = 4 * g + 2 * half;     // this lane-half's K columns/rows
      v2f a;                               // A = columns k0,k0+1 of M, row l15
      a.x = M[l15 * 16 + k0 + 0];
      a.y = M[l15 * 16 + k0 + 1];
      v2f u;                               // B = rows k0,k0+1 of U; U[k][n]=n>=k
      u.x = (l15 >= k0 + 0) ? 1.0f : 0.0f;
      u.y = (l15 >= k0 + 1) ? 1.0f : 0.0f;
      acc = __builtin_amdgcn_wmma_f32_16x16x4_f32(
          false, a, false, u, (short)0, acc, false, false);
    }
    // D layout: acc[j] = (M x U)[row = j + 8*half][col = l15]
#pragma unroll
    for (int j = 0; j < 8; ++j)
      incl[t * 256 + (j + 8 * half) * 16 + l15] = acc[j];
  }
  __syncthreads();
  if (lane == 0) {
    float run = 0.0f;
    for (int r = 0; r < 64; ++r) {         // incl[r*16+15] = row total
      carry[r] = run;
      run += incl[r * 16 + 15];
    }
    totalCnt[b] = (unsigned int)run;
  }
  __syncthreads();
  unsigned int* dst = blockOff + (size_t)b * BLOCKS_PAD;
  for (int i = lane; i < BLOCKS_PAD; i += 32) {
    float inc = incl[i] + carry[i >> 4];
    dst[i] = (unsigned int)(inc - vals[i]); // exclusive offset
  }
}

// ============================================================================
// Stable scatter: 16 in-order rounds of 256 contiguous elements; wave32
// ballot ranks + in-order cross-wave LDS scan preserve ascending flat-index
// order exactly like jnp.nonzero. Doppler gathered only for survivors
// (prefetched early -> global_prefetch_b8).
// ============================================================================
__global__ void k_scatter(const float* __restrict__ cube,
                          const float* __restrict__ dop,
                          const float* __restrict__ thr,
                          const unsigned int* __restrict__ blockOff,
                          float* __restrict__ out) {
  __shared__ unsigned int s_wcnt[8];
  __shared__ unsigned int s_woff[8];
  __shared__ unsigned int s_roundBase;
  const int b = blockIdx.y;
  const int tid = threadIdx.x;
  const int lane = tid & 31;
  const int wid = tid >> 5;
  const float t = thr[b];
  const unsigned int base = blockOff[(size_t)b * BLOCKS_PAD + blockIdx.x];
  const unsigned int chunk0 = (unsigned int)blockIdx.x * CHUNK;
  const float* csrc = cube + (size_t)b * NPER;
  const float* dsrc = dop + (size_t)b * NPER;

  float* feats = out;
  float* sind  = out + (size_t)BATCH * KPAD * 5;
  float* valid = out + (size_t)BATCH * KPAD * 9;

  if (tid == 0) s_roundBase = 0u;
  __syncthreads();

  for (int r = 0; r < 16; ++r) {
    const unsigned int gi = chunk0 + (unsigned int)(r * 256 + tid);
    const float v = csrc[gi];
    const bool pred = v > t;
    if (pred) __builtin_prefetch(&dsrc[gi], 0, 3);  // global_prefetch_b8
    const unsigned int mask = __builtin_amdgcn_ballot_w32(pred);
    const unsigned int rank =
        (unsigned int)__builtin_popcount(mask & ((1u << lane) - 1u));
    if (lane == 0) s_wcnt[wid] = (unsigned int)__builtin_popcount(mask);
    __syncthreads();
    if (tid == 0) {
      unsigned int acc = s_roundBase;
      for (int w = 0; w < 8; ++w) { s_woff[w] = acc; acc += s_wcnt[w]; }
      s_roundBase = acc;
    }
    __syncthreads();
    if (pred) {
      const unsigned int o = base + s_woff[wid] + rank;
      if (o < (unsigned int)KPAD) {
        const unsigned int row = (unsigned int)b * KPAD + o;
        const unsigned int zi = gi / (YD * XD);
        const unsigned int rem = gi % (YD * XD);
        const unsigned int yi = rem / XD;
        const unsigned int xi = rem % XD;
        feats[(size_t)row * 5 + 0] = (float)xi * (72.0f / XD);
        feats[(size_t)row * 5 + 1] = (float)yi * (32.0f / YD);
        feats[(size_t)row * 5 + 2] = (float)zi * (8.0f / ZD);
        feats[(size_t)row * 5 + 3] = v * (1.0f / POW_SCALE);
        feats[(size_t)row * 5 + 4] = dsrc[gi] - DOP_OFFSET;
        sind[(size_t)row * 4 + 0] = (float)b;
        sind[(size_t)row * 4 + 1] = (float)zi;
        sind[(size_t)row * 4 + 2] = (float)yi;
        sind[(size_t)row * 4 + 3] = (float)xi;
        valid[row] = 1.0f;
      }
    }
  }
}

// ============================================================================
extern "C" void kernel_launch(void* const* d_in, const int* in_sizes, int n_in,
                              void* d_out, int out_size, void* d_ws,
                              size_t ws_size, hipStream_t stream) {
  const float* cube = (const float*)d_in[0];
  const float* dop  = (const float*)d_in[1];
  float* out = (float*)d_out;

  // Workspace layout (uint32 units). Zero-initialized region first.
  unsigned int* ws       = (unsigned int*)d_ws;
  unsigned int* hist1    = ws;                               // B*4096
  unsigned int* hist2    = hist1 + BATCH * 4096;             // B*2*4096
  unsigned int* hist3    = hist2 + BATCH * 2 * 4096;         // B*2*256
  unsigned int* blockCnt = hist3 + BATCH * 2 * 256;          // B*1024
  unsigned int* selP1    = blockCnt + BATCH * BLOCKS_PAD;    // B*2
  unsigned int* selR1    = selP1 + BATCH * 2;                // B*2
  unsigned int* selP2    = selR1 + BATCH * 2;                // B*2
  unsigned int* selR2    = selP2 + BATCH * 2;                // B*2
  float*        thr      = (float*)(selR2 + BATCH * 2);      // B
  unsigned int* blockOff = (unsigned int*)thr + BATCH;       // B*1024
  unsigned int* totalCnt = blockOff + BATCH * BLOCKS_PAD;    // B
  const size_t ws_used_bytes =
      (size_t)((totalCnt + BATCH) - ws) * sizeof(unsigned int);

  hipMemsetAsync(d_ws, 0, ws_used_bytes, stream);
  hipMemsetAsync(d_out, 0, (size_t)out_size * sizeof(float), stream);

  const dim3 gridH(BLOCKS_PER_SAMPLE, BATCH);

  k_hist_l1<<<gridH, 256, 0, stream>>>(cube, hist1);
  k_pick_l1<<<1, 32, 0, stream>>>(hist1, selP1, selR1);
  k_hist_l2<<<gridH, 256, 0, stream>>>(cube, selP1, hist2);
  k_pick_l2<<<1, 32, 0, stream>>>(hist2, selR1, selP2, selR2);
  k_hist_l3<<<gridH, 256, 0, stream>>>(cube, selP1, selP2, hist3);
  k_pick_l3_thr<<<1, 32, 0, stream>>>(hist3, selP1, selP2, selR2, thr);
  k_count<<<gridH, 256, 0, stream>>>(cube, thr, blockCnt);
  k_scan_wmma<<<BATCH, 32, 0, stream>>>(blockCnt, blockOff, totalCnt);
  k_scatter<<<gridH, 256, 0, stream>>>(cube, dop, thr, blockOff, out);
}